// DeepStitch_49469433315386
// MI455X (gfx1250) — compile-verified
//
#include <hip/hip_runtime.h>
#include <hip/hip_bf16.h>
#include <stdint.h>

// Problem constants (from reference): B=4, C=512, H=W=128, S=16 -> K=256
#define NB   4
#define NC   512
#define NW   128
#define NHW  16384     // 128*128
#define NKP  256       // keypoints per batch

typedef __attribute__((ext_vector_type(2))) float v2f;
typedef __attribute__((ext_vector_type(8))) float v8f;

// ---------------------------------------------------------------------------
// Kernel 1: resp[b,p] = sum_c A[b,c,p];  bn[b,p] = sum_c B[b,c,p]^2
// ---------------------------------------------------------------------------
__global__ __launch_bounds__(256) void k_resp_bn(const float* __restrict__ fa,
                                                 const float* __restrict__ fb,
                                                 float* __restrict__ resp,
                                                 float* __restrict__ bn) {
    int p   = blockIdx.x * blockDim.x + threadIdx.x;   // 0 .. NB*NHW-1
    int b   = p >> 14;
    int pos = p & (NHW - 1);
    const float* ap = fa + (size_t)b * NC * NHW + pos;
    const float* bp = fb + (size_t)b * NC * NHW + pos;
    float sA = 0.f, sB = 0.f;
#pragma unroll 8
    for (int c = 0; c < NC; ++c) {
        float va = ap[(size_t)c * NHW];
        float vb = bp[(size_t)c * NHW];
        sA += va;
        sB += vb * vb;
    }
    resp[p] = sA;
    bn[p]   = sB;
}

// ---------------------------------------------------------------------------
// Kernel 2: adaptive 16x16 max-pool argmax over resp (8x8 windows).
// Strict '>' keeps the FIRST max in row-major window order (jnp.argmax).
// ---------------------------------------------------------------------------
__global__ __launch_bounds__(256) void k_pool(const float* __restrict__ resp,
                                              int* __restrict__ idxw,
                                              unsigned long long* __restrict__ packed) {
    int b  = blockIdx.x;
    int k  = threadIdx.x;            // 0..255
    int bi = k >> 4, bj = k & 15;
    int r0 = bi * 8, c0 = bj * 8;
    float best = -__builtin_inff();
    int bloc = 0;
    for (int r = 0; r < 8; ++r) {
        for (int cc = 0; cc < 8; ++cc) {
            float v = resp[b * NHW + (r0 + r) * NW + (c0 + cc)];
            if (v > best) { best = v; bloc = r * 8 + cc; }
        }
    }
    int rowA = r0 + (bloc >> 3);
    int colA = c0 + (bloc & 7);
    idxw[b * NKP + k]   = rowA * NW + colA;
    packed[b * NKP + k] = ~0ull;
}

// ---------------------------------------------------------------------------
// Kernel 3: desc[b][m][c] = A[b][c][idx[b,m]] (row-major MxK for the GEMM),
//           dn[b,m] = ||desc||^2. One wave per (b,m) pair.
// ---------------------------------------------------------------------------
__global__ __launch_bounds__(256) void k_gather(const float* __restrict__ fa,
                                                const int* __restrict__ idxw,
                                                float* __restrict__ desc,
                                                float* __restrict__ dn) {
    int gw   = blockIdx.x * 8 + (threadIdx.x >> 5);  // 0..1023 = (b,m)
    int lane = threadIdx.x & 31;
    int b = gw >> 8, m = gw & 255;
    int id = idxw[b * NKP + m];
    const float* ab = fa + (size_t)b * NC * NHW + id;
    float s = 0.f;
#pragma unroll
    for (int i = 0; i < NC / 32; ++i) {
        int c = i * 32 + lane;
        float v = ab[(size_t)c * NHW];
        desc[(size_t)gw * NC + c] = v;
        s += v * v;
    }
    for (int d = 16; d >= 1; d >>= 1) s += __shfl_xor(s, d, 32);
    if (lane == 0) dn[b * NKP + m] = s;
}

// ---------------------------------------------------------------------------
// Kernel 4: f32 WMMA GEMM  cross[b] = desc[b] (256x512) * fb[b] (512x16384),
// fused: dist = dn - 2*cross + bn, u64-packed min+argmin over N.
//
// Block = 256 threads = 8 waves = 2 M-halves x 4 N-groups; each wave owns
// 8 M-tiles x 2 N-tiles (128 acc VGPRs) so every A-fragment feeds 2 WMMAs.
// Global->LDS staging uses CDNA5 GLOBAL_LOAD_ASYNC_TO_LDS_B128 (ASYNCcnt),
// double-buffered: chunk i+1 copies overlap chunk i's 128 WMMAs per wave.
// ---------------------------------------------------------------------------
#define KCH    32                 // K-chunk
#define NCHUNK (NC / KCH)         // 16
#define AST    36                 // A row stride (144 B: 16B-aligned, conflict-free)
#define BST    136                // B row stride (544 B: 16B-aligned)
#define ABUF   (256 * AST)        // floats per A buffer
#define BBUF   (KCH * BST)        // floats per B buffer

__global__ __launch_bounds__(256) void k_gemm_min(const float* __restrict__ fb,
                                                  const float* __restrict__ desc,
                                                  const float* __restrict__ dn,
                                                  const float* __restrict__ bn,
                                                  unsigned long long* __restrict__ packed) {
    __shared__ float a_lds[2 * ABUF];   // 73728 B
    __shared__ float b_lds[2 * BBUF];   // 34816 B

    const int b    = blockIdx.x >> 7;          // 128 n-blocks per batch
    const int n0   = (blockIdx.x & 127) << 7;  // 128 columns per block
    const int t    = threadIdx.x;
    const int wave = t >> 5, lane = t & 31;
    const int half = lane >> 4, col = lane & 15;
    const int mhalf = wave & 1;                // M base 0 / 128
    const int ngrp  = wave >> 1;               // 32 columns per group
    const int cbase = ngrp * 32 + col;         // column (tile 0) within block

    float bnv0 = bn[b * NHW + n0 + cbase];
    float bnv1 = bn[b * NHW + n0 + cbase + 16];
    const float* fbb   = fb   + (size_t)b * NC * NHW;
    const float* descb = desc + (size_t)b * NKP * NC;

    // Low 32 bits of a flat LDS pointer == LDS byte offset (ISA 10.2).
    const uint32_t a_base = (uint32_t)(uintptr_t)(void*)a_lds;
    const uint32_t b_base = (uint32_t)(uintptr_t)(void*)b_lds;

    v8f acc[8][2];
    const v8f vz = {0.f, 0.f, 0.f, 0.f, 0.f, 0.f, 0.f, 0.f};
#pragma unroll
    for (int i = 0; i < 8; ++i) { acc[i][0] = vz; acc[i][1] = vz; }

    // Issue one K-chunk's async global->LDS copies (12 x B128 per thread).
    auto issue_chunk = [&](int ci) {
        const int kc = ci * KCH;
        const uint32_t abase = a_base + (uint32_t)((ci & 1) * (ABUF * 4));
        const uint32_t bbase = b_base + (uint32_t)((ci & 1) * (BBUF * 4));
#pragma unroll
        for (int j = 0; j < 8; ++j) {            // A: 256 rows x 32 floats
            int s = t + j * 256;
            int m = s >> 3, seg = s & 7;
            uint64_t ga = (uint64_t)(uintptr_t)(descb + (size_t)m * NC + kc + seg * 4);
            uint32_t la = abase + (uint32_t)(m * (AST * 4) + seg * 16);
            asm volatile("global_load_async_to_lds_b128 %0, %1, off"
                         :: "v"(la), "v"(ga) : "memory");
        }
#pragma unroll
        for (int j = 0; j < 4; ++j) {            // B: 32 rows x 128 floats
            int s = t + j * 256;
            int r = s >> 5, seg = s & 31;
            uint64_t ga = (uint64_t)(uintptr_t)(fbb + (size_t)(kc + r) * NHW + n0 + seg * 4);
            uint32_t la = bbase + (uint32_t)(r * (BST * 4) + seg * 16);
            asm volatile("global_load_async_to_lds_b128 %0, %1, off"
                         :: "v"(la), "v"(ga) : "memory");
        }
    };

    issue_chunk(0);
    for (int ci = 0; ci < NCHUNK; ++ci) {
        if (ci + 1 < NCHUNK) {
            issue_chunk(ci + 1);   // into the other buffer (free since ci-1's barrier)
            // In-order completion: <=12 outstanding => all of chunk ci landed.
            asm volatile("s_wait_asynccnt 0xc" ::: "memory");
        } else {
            asm volatile("s_wait_asynccnt 0x0" ::: "memory");
        }
        __syncthreads();           // all waves' chunk-ci data visible in LDS

        const float* al = a_lds + (ci & 1) * ABUF;
        const float* bl = b_lds + (ci & 1) * BBUF;
        for (int kk = 0; kk < KCH; kk += 4) {
            // f32 16x16x4: lanes<16 -> K=kk,kk+1; lanes>=16 -> K=kk+2,kk+3
            const int rb = kk + 2 * half;
            v2f bf0, bf1;
            bf0.x = bl[rb * BST + cbase];        bf0.y = bl[(rb + 1) * BST + cbase];
            bf1.x = bl[rb * BST + cbase + 16];   bf1.y = bl[(rb + 1) * BST + cbase + 16];
#pragma unroll
            for (int mt = 0; mt < 8; ++mt) {
                const int m = mhalf * 128 + mt * 16 + col;
                v2f af = *(const v2f*)&al[m * AST + rb];
                acc[mt][0] = __builtin_amdgcn_wmma_f32_16x16x4_f32(
                    false, af, false, bf0, (short)0, acc[mt][0], false, false);
                acc[mt][1] = __builtin_amdgcn_wmma_f32_16x16x4_f32(
                    false, af, false, bf1, (short)0, acc[mt][1], false, false);
            }
        }
        __syncthreads();           // done reading this buffer; reusable
    }

    // Epilogue: D layout -> VGPR g, lanes<16: M=+g, lanes>=16: M=+g+8; N=col.
    // Pack (orderable float key << 32 | n), min-reduce within 16-lane halves,
    // then global u64 atomicMin (tie -> smallest n, matching jnp.argmin).
#pragma unroll 2
    for (int mt = 0; mt < 8; ++mt) {
#pragma unroll
        for (int g = 0; g < 8; ++g) {
            const int m = mhalf * 128 + mt * 16 + g + 8 * half;
            const float dnv = dn[b * NKP + m];
#pragma unroll
            for (int nt = 0; nt < 2; ++nt) {
                int n = n0 + ngrp * 32 + nt * 16 + col;
                float dist = dnv - 2.0f * acc[mt][nt] [g] + (nt ? bnv1 : bnv0);
                unsigned int fbits = __float_as_uint(dist);
                unsigned int key = (fbits & 0x80000000u) ? ~fbits : (fbits | 0x80000000u);
                unsigned long long p = ((unsigned long long)key << 32) | (unsigned int)n;
                for (int s = 1; s < 16; s <<= 1) {
                    unsigned long long q = __shfl_xor(p, s, 32);
                    if (q < p) p = q;
                }
                if (col == 0) atomicMin(&packed[b * NKP + m], p);
            }
        }
    }
}

// ---------------------------------------------------------------------------
// Kernel 5: decode min vals, compute (drow,dcol) mode per batch.
// out[0..7] = offsets (B x 2, as float), out[8..8+1023] = min_vals (B x K).
// ---------------------------------------------------------------------------
__global__ __launch_bounds__(256) void k_finalize(const unsigned long long* __restrict__ packed,
                                                  const int* __restrict__ idxw,
                                                  float* __restrict__ out) {
    __shared__ int s_code[256], s_dr[256], s_dc[256];
    __shared__ unsigned int s_red[256];
    int b = blockIdx.x, k = threadIdx.x;

    unsigned long long p = packed[b * NKP + k];
    unsigned int key    = (unsigned int)(p >> 32);
    unsigned int minidx = (unsigned int)(p & 0xFFFFFFFFu);
    unsigned int fbits  = (key & 0x80000000u) ? (key & 0x7FFFFFFFu) : ~key;
    out[8 + b * NKP + k] = __uint_as_float(fbits);   // min_vals

    int idA  = idxw[b * NKP + k];
    int rowA = idA >> 7,            colA = idA & (NW - 1);
    int rowB = (int)(minidx >> 7),  colB = (int)(minidx & (NW - 1));
    int dr = rowA - rowB, dc = colA - colB;
    s_code[k] = dr * (4 * NW) + dc;
    s_dr[k] = dr; s_dc[k] = dc;
    __syncthreads();

    int my = s_code[k], cnt = 0;
    for (int j = 0; j < NKP; ++j) cnt += (s_code[j] == my);
    // argmax of counts, tie -> smallest k (first occurrence, like jnp.argmax)
    s_red[k] = ((unsigned int)cnt << 8) | (unsigned int)(255 - k);
    __syncthreads();
    for (int s = 128; s >= 1; s >>= 1) {
        if (k < s) {
            unsigned int a = s_red[k], c2 = s_red[k + s];
            s_red[k] = (c2 > a) ? c2 : a;
        }
        __syncthreads();
    }
    if (k == 0) {
        int bk = 255 - (int)(s_red[0] & 255u);
        out[b * 2 + 0] = (float)s_dr[bk];
        out[b * 2 + 1] = (float)s_dc[bk];
    }
}

// ---------------------------------------------------------------------------
extern "C" void kernel_launch(void* const* d_in, const int* in_sizes, int n_in,
                              void* d_out, int out_size, void* d_ws, size_t ws_size,
                              hipStream_t stream) {
    const float* fa = (const float*)d_in[0];
    const float* fb = (const float*)d_in[1];
    float* out = (float*)d_out;

    // Workspace layout (bytes), total ~2.52 MB
    char* ws = (char*)d_ws;
    float* resp               = (float*)(ws + 0);          // 256 KB
    float* bn                 = (float*)(ws + 262144);     // 256 KB
    float* dn                 = (float*)(ws + 524288);     // 4 KB
    int*   idxw               = (int*)  (ws + 528384);     // 4 KB
    unsigned long long* packd = (unsigned long long*)(ws + 532480); // 8 KB (8B aligned)
    float* desc               = (float*)(ws + 540672);     // 2 MB

    k_resp_bn <<<NB * NHW / 256, 256, 0, stream>>>(fa, fb, resp, bn);
    k_pool    <<<NB,             256, 0, stream>>>(resp, idxw, packd);
    k_gather  <<<NB * NKP / 8,   256, 0, stream>>>(fa, idxw, desc, dn);
    k_gemm_min<<<NB * (NHW/128), 256, 0, stream>>>(fb, desc, dn, bn, packd);
    k_finalize<<<NB,             256, 0, stream>>>(packd, idxw, out);
}